// MyModel_87522843560033
// MI455X (gfx1250) — compile-verified
//
#include <hip/hip_runtime.h>
#include <hip/hip_bf16.h>
#include <math.h>

typedef __attribute__((ext_vector_type(2))) float v2f;
typedef __attribute__((ext_vector_type(4))) float v4f;
typedef __attribute__((ext_vector_type(8))) float v8f;

#define B_   32
#define T_   1024
#define F_   40
#define H_   256
#define G_   1024   // 4*H
#define TP_  342    // ceil(1024/3), maxpool3 SAME output length
#define VO_  30     // V+1

// d_out layout (floats): ctc_s, len_s, ctc_c, len_c, close
#define OUT_CTC_S 0
#define OUT_LEN_S (B_*TP_*VO_)
#define OUT_CTC_C (OUT_LEN_S + B_)
#define OUT_LEN_C (OUT_CTC_C + B_*TP_*VO_)
#define OUT_CLOSE (OUT_LEN_C + B_)

// ---------------------------------------------------------------------------
// Exact-fp32 WMMA: D = A(16x4) * B(4x16) + C, via V_WMMA_F32_16X16X4_F32.
// A per-lane: m = lane&15, k = 2*(lane>>4) + e
// B per-lane: n = lane&15, k = 2*(lane>>4) + e
// C/D per-lane: n = lane&15, m = 8*(lane>>4) + v
// ---------------------------------------------------------------------------
__device__ __forceinline__ v8f wmma4(v2f a, v2f b, v8f c) {
  return __builtin_amdgcn_wmma_f32_16x16x4_f32(false, a, false, b, (short)0, c,
                                               false, false);
}

// XOR-swizzled LDS addressing: kills the 16-way bank conflict of stride-256
// rows while keeping 8-byte pairs contiguous (xor value is even).
__device__ __forceinline__ int swz(int row, int col) {
  return row * H_ + (col ^ ((row & 15) << 1));
}

// ---------------------------------------------------------------------------
// Pre-swizzle combined weights [Wk ; Wr] (K x 1024) into B-fragment layout.
// Fragment q = gate*16 + unit_tile; two k-tiles packed per float4 so the
// recurrent loop does one global_load_b128 per lane per 2 k-tiles.
// out[((q*KT2 + kt2)*32 + lane)*4 + e], KT2 = K/8.
// ---------------------------------------------------------------------------
__global__ void prep_w_kernel(const float* __restrict__ wk,
                              const float* __restrict__ wr,
                              int kx, int kt2n, float* __restrict__ out, int n) {
  int i = blockIdx.x * blockDim.x + threadIdx.x;
  if (i >= n) return;
  int e = i & 3, lane = (i >> 2) & 31, kt2 = (i >> 7) % kt2n, q = (i >> 7) / kt2n;
  int kt  = kt2 * 2 + (e >> 1);
  int k   = kt * 4 + 2 * (lane >> 4) + (e & 1);
  int col = (q >> 4) * H_ + (q & 15) * 16 + (lane & 15);
  out[i] = (k < kx) ? wk[(size_t)k * G_ + col] : wr[(size_t)(k - kx) * G_ + col];
}

// Pre-clip x and swizzle into A-fragment layout (K = F_ = 40 = 10 k-tiles).
// xf[(((mt*T + t)*5 + kt2)*32 + lane)*4 + e]
__global__ void prep_x_kernel(const float* __restrict__ x, float* __restrict__ xf) {
  int i = blockIdx.x * blockDim.x + threadIdx.x;
  const int n = 2 * T_ * 5 * 128;
  if (i >= n) return;
  int e = i & 3, lane = (i >> 2) & 31, kt2 = (i >> 7) % 5;
  int t = (i >> 7) / 5 % T_, mt = i / (128 * 5 * T_);
  int kt = kt2 * 2 + (e >> 1);
  int k  = kt * 4 + 2 * (lane >> 4) + (e & 1);
  int m  = mt * 16 + (lane & 15);
  float v = x[((size_t)m * T_ + t) * F_ + k];
  xf[i] = fminf(3.0f, fmaxf(-3.0f, v));
}

// ---------------------------------------------------------------------------
// LSTM0 (40 -> 256) with fused clip (in prep) + fused maxpool3 SAME.
// One persistent 32-wave workgroup per branch.  Wave (ut, mt) owns all four
// gate tiles of 16 units x 16 batches; cell state lives in registers for all
// 1024 steps; h broadcast through swizzled LDS; weights streamed from L2
// every step.  A single opaque zero offset (redefined by empty asm each
// iteration) keeps LICM from hoisting the weight stream while preserving the
// global address space of the loads.
// ---------------------------------------------------------------------------
__global__ void __launch_bounds__(1024, 1)
lstm0_kernel(const float* __restrict__ Xf,
             const float* __restrict__ WfS, const float* __restrict__ WfC,
             const float* __restrict__ bS,  const float* __restrict__ bC,
             const int* __restrict__ xlen, float* __restrict__ pooled) {
  const float* Wf   = blockIdx.x ? WfC : WfS;
  const float* bias = blockIdx.x ? bC  : bS;
  pooled += (size_t)blockIdx.x * (B_ * TP_ * H_);

  __shared__ float hbuf[B_ * H_];
  const int tid = threadIdx.x, lane = tid & 31, w = tid >> 5;
  const int ut = w & 15, mt = w >> 4;
  const int half = lane >> 4, ln = lane & 15;
  const int j = ut * 16 + ln;
  const int arow = mt * 16 + ln;   // A-fragment row
  const int lofs = lane * 4;       // fragment lane offset (floats)

  float bg[4];
#pragma unroll
  for (int g = 0; g < 4; ++g) bg[g] = bias[g * H_ + j];
  int lens[8];
#pragma unroll
  for (int v = 0; v < 8; ++v) lens[v] = xlen[mt * 16 + half * 8 + v];

  for (int i = tid; i < B_ * H_; i += 1024) hbuf[i] = 0.0f;
  __syncthreads();

  const float* wf0 = Wf + (size_t)(0 * 16 + ut) * 37 * 128 + lofs;
  const float* wf1 = Wf + (size_t)(1 * 16 + ut) * 37 * 128 + lofs;
  const float* wf2 = Wf + (size_t)(2 * 16 + ut) * 37 * 128 + lofs;
  const float* wf3 = Wf + (size_t)(3 * 16 + ut) * 37 * 128 + lofs;
  const float* xp  = Xf + (size_t)(mt * T_) * 640 + lofs;  // advanced by 640/t

  v8f cst = (v8f)0.0f, hst = (v8f)0.0f, pmax = (v8f)(-3.0e38f);

  for (int t = 0; t < T_; ++t) {
    // Opaque zero: blinds LICM so the weight stream stays in the step loop,
    // without dirtying the (global) pointer chain.
    int zoff = 0;
    asm volatile("" : "+v"(zoff));

    v8f acc[4];
#pragma unroll
    for (int g = 0; g < 4; ++g) acc[g] = (v8f)0.0f;

    // ---- x part: k-tiles 0..9 (K=40), A from pre-swizzled Xf (coalesced) ----
    for (int kt2 = 0; kt2 < 5; ++kt2) {
      v4f av = *(const v4f*)(xp + kt2 * 128);
      v2f a0 = {av[0], av[1]}, a1 = {av[2], av[3]};
      v4f w0v = *(const v4f*)(wf0 + zoff + kt2 * 128);
      v4f w1v = *(const v4f*)(wf1 + zoff + kt2 * 128);
      v4f w2v = *(const v4f*)(wf2 + zoff + kt2 * 128);
      v4f w3v = *(const v4f*)(wf3 + zoff + kt2 * 128);
      acc[0] = wmma4(a0, (v2f){w0v[0], w0v[1]}, acc[0]);
      acc[0] = wmma4(a1, (v2f){w0v[2], w0v[3]}, acc[0]);
      acc[1] = wmma4(a0, (v2f){w1v[0], w1v[1]}, acc[1]);
      acc[1] = wmma4(a1, (v2f){w1v[2], w1v[3]}, acc[1]);
      acc[2] = wmma4(a0, (v2f){w2v[0], w2v[1]}, acc[2]);
      acc[2] = wmma4(a1, (v2f){w2v[2], w2v[3]}, acc[2]);
      acc[3] = wmma4(a0, (v2f){w3v[0], w3v[1]}, acc[3]);
      acc[3] = wmma4(a1, (v2f){w3v[2], w3v[3]}, acc[3]);
    }
    // ---- h part: k-tiles 10..73 (k=40..295), A from swizzled LDS ----
    for (int kt2 = 5; kt2 < 37; ++kt2) {
      int koff = kt2 * 8 - 40 + 2 * half;
      v2f a0 = *(const v2f*)(hbuf + swz(arow, koff));
      v2f a1 = *(const v2f*)(hbuf + swz(arow, koff + 4));
      v4f w0v = *(const v4f*)(wf0 + zoff + kt2 * 128);
      v4f w1v = *(const v4f*)(wf1 + zoff + kt2 * 128);
      v4f w2v = *(const v4f*)(wf2 + zoff + kt2 * 128);
      v4f w3v = *(const v4f*)(wf3 + zoff + kt2 * 128);
      acc[0] = wmma4(a0, (v2f){w0v[0], w0v[1]}, acc[0]);
      acc[0] = wmma4(a1, (v2f){w0v[2], w0v[3]}, acc[0]);
      acc[1] = wmma4(a0, (v2f){w1v[0], w1v[1]}, acc[1]);
      acc[1] = wmma4(a1, (v2f){w1v[2], w1v[3]}, acc[1]);
      acc[2] = wmma4(a0, (v2f){w2v[0], w2v[1]}, acc[2]);
      acc[2] = wmma4(a1, (v2f){w2v[2], w2v[3]}, acc[2]);
      acc[3] = wmma4(a0, (v2f){w3v[0], w3v[1]}, acc[3]);
      acc[3] = wmma4(a1, (v2f){w3v[2], w3v[3]}, acc[3]);
    }
    xp += 640;  // strength-reduced Xf walk (no per-step mul, no spill)

    // ---- gates + cell update (registers only) ----
#pragma unroll
    for (int v = 0; v < 8; ++v) {
      float zi = acc[0][v] + bg[0];
      float zf = acc[1][v] + bg[1];
      float zg = acc[2][v] + bg[2];
      float zo = acc[3][v] + bg[3];
      float ig = 1.0f / (1.0f + expf(-zi));
      float fg = 1.0f / (1.0f + expf(-zf));
      float gg = tanhf(zg);
      float og = 1.0f / (1.0f + expf(-zo));
      float cn = fg * cst[v] + ig * gg;
      float hn = og * tanhf(cn);
      bool m = t < lens[v];
      cst[v] = m ? cn : cst[v];
      hst[v] = m ? hn : hst[v];
    }

    __syncthreads();  // all reads of h_{t-1} complete
#pragma unroll
    for (int v = 0; v < 8; ++v)
      hbuf[swz(mt * 16 + half * 8 + v, j)] = hst[v];

    // fused maxpool3 SAME: window o=(t+1)/3 closes at t%3==1 or t==T-1
#pragma unroll
    for (int v = 0; v < 8; ++v) pmax[v] = fmaxf(pmax[v], hst[v]);
    if ((t % 3) == 1 || t == T_ - 1) {
      int o = (t + 1) / 3;
#pragma unroll
      for (int v = 0; v < 8; ++v)
        pooled[((size_t)(mt * 16 + half * 8 + v) * TP_ + o) * H_ + j] = pmax[v];
      pmax = (v8f)(-3.0e38f);
    }
    __syncthreads();  // h_t visible before next step
  }
}

// ---------------------------------------------------------------------------
// LSTM1 (256 -> 256) over T'=342; K = 512.  u_t staged coalesced into LDS,
// recurrent h in LDS, weights streamed from L2 per step.
// ---------------------------------------------------------------------------
__global__ void __launch_bounds__(1024, 1)
lstm1_kernel(const float* __restrict__ U,
             const float* __restrict__ WfS, const float* __restrict__ WfC,
             const float* __restrict__ bS,  const float* __restrict__ bC,
             const int* __restrict__ xlen, float* __restrict__ Ys) {
  const float* Wf   = blockIdx.x ? WfC : WfS;
  const float* bias = blockIdx.x ? bC  : bS;
  U  += (size_t)blockIdx.x * (B_ * TP_ * H_);
  Ys += (size_t)blockIdx.x * (B_ * TP_ * H_);

  __shared__ float hbuf[B_ * H_];
  __shared__ float ubuf[B_ * H_];
  const int tid = threadIdx.x, lane = tid & 31, w = tid >> 5;
  const int ut = w & 15, mt = w >> 4;
  const int half = lane >> 4, ln = lane & 15;
  const int j = ut * 16 + ln;
  const int arow = mt * 16 + ln;
  const int lofs = lane * 4;

  float bg[4];
#pragma unroll
  for (int g = 0; g < 4; ++g) bg[g] = bias[g * H_ + j];
  int lens[8];
#pragma unroll
  for (int v = 0; v < 8; ++v)
    lens[v] = (xlen[mt * 16 + half * 8 + v] + 2) / 3;  // seq_len after pool

  for (int i = tid; i < B_ * H_; i += 1024) hbuf[i] = 0.0f;
  __syncthreads();

  const float* wf0 = Wf + (size_t)(0 * 16 + ut) * 64 * 128 + lofs;
  const float* wf1 = Wf + (size_t)(1 * 16 + ut) * 64 * 128 + lofs;
  const float* wf2 = Wf + (size_t)(2 * 16 + ut) * 64 * 128 + lofs;
  const float* wf3 = Wf + (size_t)(3 * 16 + ut) * 64 * 128 + lofs;

  v8f cst = (v8f)0.0f, hst = (v8f)0.0f;

  for (int t = 0; t < TP_; ++t) {
    int zoff = 0;
    asm volatile("" : "+v"(zoff));

    // ---- stage u_t into LDS, fully coalesced (each row contiguous 1KB) ----
    for (int i = tid; i < B_ * H_; i += 1024) {
      int r = i >> 8, c = i & 255;
      ubuf[swz(r, c)] = U[((size_t)r * TP_ + t) * H_ + c];
    }
    __syncthreads();

    v8f acc[4];
#pragma unroll
    for (int g = 0; g < 4; ++g) acc[g] = (v8f)0.0f;

    // ---- input part: k-tiles 0..63, A from LDS ----
    for (int kt2 = 0; kt2 < 32; ++kt2) {
      int koff = kt2 * 8 + 2 * half;
      v2f a0 = *(const v2f*)(ubuf + swz(arow, koff));
      v2f a1 = *(const v2f*)(ubuf + swz(arow, koff + 4));
      v4f w0v = *(const v4f*)(wf0 + zoff + kt2 * 128);
      v4f w1v = *(const v4f*)(wf1 + zoff + kt2 * 128);
      v4f w2v = *(const v4f*)(wf2 + zoff + kt2 * 128);
      v4f w3v = *(const v4f*)(wf3 + zoff + kt2 * 128);
      acc[0] = wmma4(a0, (v2f){w0v[0], w0v[1]}, acc[0]);
      acc[0] = wmma4(a1, (v2f){w0v[2], w0v[3]}, acc[0]);
      acc[1] = wmma4(a0, (v2f){w1v[0], w1v[1]}, acc[1]);
      acc[1] = wmma4(a1, (v2f){w1v[2], w1v[3]}, acc[1]);
      acc[2] = wmma4(a0, (v2f){w2v[0], w2v[1]}, acc[2]);
      acc[2] = wmma4(a1, (v2f){w2v[2], w2v[3]}, acc[2]);
      acc[3] = wmma4(a0, (v2f){w3v[0], w3v[1]}, acc[3]);
      acc[3] = wmma4(a1, (v2f){w3v[2], w3v[3]}, acc[3]);
    }
    // ---- recurrent part: k-tiles 64..127, A from LDS ----
    for (int kt2 = 32; kt2 < 64; ++kt2) {
      int koff = kt2 * 8 - 256 + 2 * half;
      v2f a0 = *(const v2f*)(hbuf + swz(arow, koff));
      v2f a1 = *(const v2f*)(hbuf + swz(arow, koff + 4));
      v4f w0v = *(const v4f*)(wf0 + zoff + kt2 * 128);
      v4f w1v = *(const v4f*)(wf1 + zoff + kt2 * 128);
      v4f w2v = *(const v4f*)(wf2 + zoff + kt2 * 128);
      v4f w3v = *(const v4f*)(wf3 + zoff + kt2 * 128);
      acc[0] = wmma4(a0, (v2f){w0v[0], w0v[1]}, acc[0]);
      acc[0] = wmma4(a1, (v2f){w0v[2], w0v[3]}, acc[0]);
      acc[1] = wmma4(a0, (v2f){w1v[0], w1v[1]}, acc[1]);
      acc[1] = wmma4(a1, (v2f){w1v[2], w1v[3]}, acc[1]);
      acc[2] = wmma4(a0, (v2f){w2v[0], w2v[1]}, acc[2]);
      acc[2] = wmma4(a1, (v2f){w2v[2], w2v[3]}, acc[2]);
      acc[3] = wmma4(a0, (v2f){w3v[0], w3v[1]}, acc[3]);
      acc[3] = wmma4(a1, (v2f){w3v[2], w3v[3]}, acc[3]);
    }

#pragma unroll
    for (int v = 0; v < 8; ++v) {
      float zi = acc[0][v] + bg[0];
      float zf = acc[1][v] + bg[1];
      float zg = acc[2][v] + bg[2];
      float zo = acc[3][v] + bg[3];
      float ig = 1.0f / (1.0f + expf(-zi));
      float fg = 1.0f / (1.0f + expf(-zf));
      float gg = tanhf(zg);
      float og = 1.0f / (1.0f + expf(-zo));
      float cn = fg * cst[v] + ig * gg;
      float hn = og * tanhf(cn);
      bool m = t < lens[v];
      cst[v] = m ? cn : cst[v];
      hst[v] = m ? hn : hst[v];
    }

    __syncthreads();  // reads of h_{t-1}/u_t complete
#pragma unroll
    for (int v = 0; v < 8; ++v) {
      hbuf[swz(mt * 16 + half * 8 + v, j)] = hst[v];
      Ys[((size_t)(mt * 16 + half * 8 + v) * TP_ + t) * H_ + j] = hst[v];
    }
    __syncthreads();  // h_t visible before next step
  }
}

// CTC projection: out[b,t,v] = ys2[b,t,:] @ ctc_w[:,v] + ctc_b[v], both branches.
// ctc_w staged in LDS per block.
__global__ void ctc_kernel(const float* __restrict__ ys2,
                           const float* __restrict__ cw,
                           const float* __restrict__ cb,
                           float* __restrict__ out) {
  __shared__ float cws[H_ * VO_];
  for (int i = threadIdx.x; i < H_ * VO_; i += blockDim.x) cws[i] = cw[i];
  __syncthreads();

  int i = blockIdx.x * blockDim.x + threadIdx.x;
  const int per = B_ * TP_ * VO_;
  if (i >= 2 * per) return;
  int br = i / per, r = i % per;
  int v = r % VO_, bt = r / VO_;
  const float* ys = ys2 + (size_t)br * (B_ * TP_ * H_) + (size_t)bt * H_;
  float s = cb[v];
  for (int k = 0; k < H_; ++k) s = fmaf(ys[k], cws[k * VO_ + v], s);
  out[(br ? OUT_CTC_C : OUT_CTC_S) + r] = s;
}

// Lengths + per-batch closeness flag.
__global__ void finalize_kernel(float* __restrict__ out, const int* __restrict__ xlen) {
  __shared__ int ok;
  int b = blockIdx.x, tid = threadIdx.x;
  if (tid == 0) ok = 1;
  __syncthreads();
  int good = 1;
  const int n = TP_ * VO_;
  for (int i = tid; i < n; i += blockDim.x) {
    float a = out[OUT_CTC_S + b * n + i];
    float c = out[OUT_CTC_C + b * n + i];
    if (!(fabsf(a - c) < 1e-5f)) good = 0;
  }
  if (!good) atomicAnd(&ok, 0);
  __syncthreads();
  if (tid == 0) {
    float sl = (float)((xlen[b] + 2) / 3);
    out[OUT_LEN_S + b] = sl;
    out[OUT_LEN_C + b] = sl;
    out[OUT_CLOSE + b] = ok ? 1.0f : 0.0f;
  }
}

extern "C" void kernel_launch(void* const* d_in, const int* in_sizes, int n_in,
                              void* d_out, int out_size, void* d_ws, size_t ws_size,
                              hipStream_t stream) {
  (void)in_sizes; (void)n_in; (void)out_size; (void)ws_size;
  const float* x     = (const float*)d_in[0];
  const int*   xlen  = (const int*)d_in[1];
  const float* ws0_k = (const float*)d_in[2];
  const float* ws0_r = (const float*)d_in[3];
  const float* ws0_b = (const float*)d_in[4];
  const float* ws1_k = (const float*)d_in[5];
  const float* ws1_r = (const float*)d_in[6];
  const float* ws1_b = (const float*)d_in[7];
  const float* wc0_k = (const float*)d_in[8];
  const float* wc0_r = (const float*)d_in[9];
  const float* wc0_b = (const float*)d_in[10];
  const float* wc1_k = (const float*)d_in[11];
  const float* wc1_r = (const float*)d_in[12];
  const float* wc1_b = (const float*)d_in[13];
  const float* ctc_w = (const float*)d_in[14];
  const float* ctc_b = (const float*)d_in[15];
  float* out = (float*)d_out;
  float* ws  = (float*)d_ws;

  const size_t NXF  = (size_t)2 * T_ * 5 * 128;   // x fragments
  const size_t NW0  = (size_t)64 * 37 * 128;      // 296x1024 combined
  const size_t NW1  = (size_t)64 * 64 * 128;      // 512x1024 combined
  const size_t NACT = (size_t)2 * B_ * TP_ * H_;  // both branches

  float* Xf     = ws;  ws += NXF;
  float* WfS0   = ws;  ws += NW0;
  float* WfC0   = ws;  ws += NW0;
  float* WfS1   = ws;  ws += NW1;
  float* WfC1   = ws;  ws += NW1;
  float* pooled = ws;  ws += NACT;   // reused as ys2 after run a consumes it
  float* ys1    = ws;  ws += NACT;
  float* ys2    = pooled;

  prep_w_kernel<<<(int)((NW0 + 255) / 256), 256, 0, stream>>>(ws0_k, ws0_r, 40, 37, WfS0, (int)NW0);
  prep_w_kernel<<<(int)((NW0 + 255) / 256), 256, 0, stream>>>(wc0_k, wc0_r, 40, 37, WfC0, (int)NW0);
  prep_w_kernel<<<(int)((NW1 + 255) / 256), 256, 0, stream>>>(ws1_k, ws1_r, 256, 64, WfS1, (int)NW1);
  prep_w_kernel<<<(int)((NW1 + 255) / 256), 256, 0, stream>>>(wc1_k, wc1_r, 256, 64, WfC1, (int)NW1);
  prep_x_kernel<<<(int)((NXF + 255) / 256), 256, 0, stream>>>(x, Xf);

  lstm0_kernel<<<2, 1024, 0, stream>>>(Xf, WfS0, WfC0, ws0_b, wc0_b, xlen, pooled);
  lstm1_kernel<<<2, 1024, 0, stream>>>(pooled, WfS1, WfC1, ws1_b, wc1_b, xlen, ys1);
  lstm1_kernel<<<2, 1024, 0, stream>>>(ys1,    WfS1, WfC1, ws1_b, wc1_b, xlen, ys2);

  const int nctc = 2 * B_ * TP_ * VO_;
  ctc_kernel<<<(nctc + 255) / 256, 256, 0, stream>>>(ys2, ctc_w, ctc_b, out);
  finalize_kernel<<<B_, 256, 0, stream>>>(out, xlen);
}